// IAFLayer_65824668778978
// MI455X (gfx1250) — compile-verified
//
#include <hip/hip_runtime.h>

typedef __attribute__((ext_vector_type(2))) float v2f;
typedef __attribute__((ext_vector_type(8))) float v8f;

#define DD 128

// One wave (32 lanes) per 16 batch rows.
// Blocked autoregressive recurrence:
//   cross-block (k < 16*I) contributions via V_WMMA_F32_16X16X4_F32 fp32 GEMMs,
//   intra-block 16 sequential steps on lanes 0..15 (one batch row per lane).
__global__ __launch_bounds__(32) void iaf_layer_kernel(
    const float* __restrict__ x,
    const float* __restrict__ W1,  const float* __restrict__ b1,
    const float* __restrict__ Wmu, const float* __restrict__ bmu,
    const float* __restrict__ Wls, const float* __restrict__ bls,
    float* __restrict__ out, int B)
{
    __shared__ float Zh[16 * DD];   // finalized z history, row-major [16][128]
    __shared__ float Hh[16 * DD];   // finalized h history, row-major [16][128]
    __shared__ float tH[256], tM[256], tL[256];   // C-tile transpose bounce
    __shared__ float wd1[256], wdm[256], wdl[256]; // diagonal 16x16 W blocks

    const int lane  = threadIdx.x;          // 0..31
    const int r0    = blockIdx.x * 16;      // batch row base of this tile
    const int mrow  = lane & 15;            // row (A) / col (B) index for WMMA operands
    const int khalf = (lane >> 4) * 2;      // K sub-offset per ISA 16x4 fp32 layout

    float ldj = 0.0f;                       // lanes 0..15: log-det-jacobian of row r0+lane

    for (int I = 0; I < 8; ++I) {
        const int c0 = I * 16;

        // Stage the strict-lower diagonal 16x16 blocks of each weight matrix in LDS.
        {
            const int e0 = lane * 8;
            #pragma unroll
            for (int q = 0; q < 8; ++q) {
                const int e  = e0 + q;
                const int t2 = e >> 4, t = e & 15;
                wd1[e] = W1 [(c0 + t2) * DD + c0 + t];
                wdm[e] = Wmu[(c0 + t2) * DD + c0 + t];
                wdl[e] = Wls[(c0 + t2) * DD + c0 + t];
            }
        }

        // Cross-block GEMMs: C[m][n] = sum_{k<16I} A[m][k] * W[c0+n][k]
        v8f cH = {0.f,0.f,0.f,0.f,0.f,0.f,0.f,0.f};
        v8f cM = {0.f,0.f,0.f,0.f,0.f,0.f,0.f,0.f};
        v8f cL = {0.f,0.f,0.f,0.f,0.f,0.f,0.f,0.f};
        const int nkt = 4 * I;              // K tiles of 4 (uniform trip count)
        for (int kt = 0; kt < nkt; ++kt) {
            const int k0 = kt * 4;
            // A operands (16x4 fp32): lane L holds row m=L&15, K pair khalf..khalf+1
            v2f aZ = *(const v2f*)&Zh[mrow * DD + k0 + khalf];
            v2f aH = *(const v2f*)&Hh[mrow * DD + k0 + khalf];
            // B operands (4x16 fp32): lane L holds col n=L&15, same K pair (W transposed)
            v2f bW1 = *(const v2f*)&W1 [(c0 + mrow) * DD + k0 + khalf];
            v2f bWm = *(const v2f*)&Wmu[(c0 + mrow) * DD + k0 + khalf];
            v2f bWl = *(const v2f*)&Wls[(c0 + mrow) * DD + k0 + khalf];
            cH = __builtin_amdgcn_wmma_f32_16x16x4_f32(false, aZ, false, bW1, (short)0, cH, false, false);
            cM = __builtin_amdgcn_wmma_f32_16x16x4_f32(false, aH, false, bWm, (short)0, cM, false, false);
            cL = __builtin_amdgcn_wmma_f32_16x16x4_f32(false, aH, false, bWl, (short)0, cL, false, false);
        }

        // Spill C tiles (ISA C-layout) to LDS row-major: lane L<16 -> rows 0..7 of col L,
        // lane L>=16 -> rows 8..15 of col L-16.
        {
            const int mb = (lane >> 4) * 8;
            const int n  = lane & 15;
            #pragma unroll
            for (int r = 0; r < 8; ++r) {
                tH[(mb + r) * 16 + n] = cH[r];
                tM[(mb + r) * 16 + n] = cM[r];
                tL[(mb + r) * 16 + n] = cL[r];
            }
        }
        __syncthreads();

        // Sequential intra-block phase: lanes 0..15, one batch row each.
        if (lane < 16) {
            const int r = lane;
            float Hrow[16], Mrow[16], Lrow[16], xr[16];
            #pragma unroll
            for (int t = 0; t < 16; ++t) {
                Hrow[t] = tH[r * 16 + t] + b1 [c0 + t];
                Mrow[t] = tM[r * 16 + t] + bmu[c0 + t];
                Lrow[t] = tL[r * 16 + t] + bls[c0 + t];
                xr[t]   = x[(size_t)(r0 + r) * DD + c0 + t];
            }
            float zr[16], hr[16];
            #pragma unroll
            for (int t = 0; t < 16; ++t) {
                const float h  = fmaxf(Hrow[t], 0.0f);   // h[:, c0+t] finalized
                const float ls = Lrow[t];
                const float zv = Mrow[t] + xr[t] * __expf(ls);
                ldj += ls;
                zr[t] = zv;
                hr[t] = h;
                #pragma unroll
                for (int t2 = t + 1; t2 < 16; ++t2) {
                    Hrow[t2] = fmaf(zv, wd1[t2 * 16 + t], Hrow[t2]);
                    Mrow[t2] = fmaf(h,  wdm[t2 * 16 + t], Mrow[t2]);
                    Lrow[t2] = fmaf(h,  wdl[t2 * 16 + t], Lrow[t2]);
                }
            }
            // Publish finalized z / h history for later blocks' GEMMs; emit z output.
            #pragma unroll
            for (int t = 0; t < 16; ++t) {
                Zh[r * DD + c0 + t] = zr[t];
                Hh[r * DD + c0 + t] = hr[t];
                out[(size_t)(r0 + r) * DD + c0 + t] = zr[t];
            }
        }
        __syncthreads();
    }

    if (lane < 16) {
        out[(size_t)B * DD + r0 + lane] = ldj;   // ldj appended after z
    }
}

extern "C" void kernel_launch(void* const* d_in, const int* in_sizes, int n_in,
                              void* d_out, int out_size, void* d_ws, size_t ws_size,
                              hipStream_t stream) {
    (void)n_in; (void)out_size; (void)d_ws; (void)ws_size;
    const float* x   = (const float*)d_in[0];
    const float* W1  = (const float*)d_in[1];
    const float* b1  = (const float*)d_in[2];
    const float* Wmu = (const float*)d_in[3];
    const float* bmu = (const float*)d_in[4];
    const float* Wls = (const float*)d_in[5];
    const float* bls = (const float*)d_in[6];
    float* out = (float*)d_out;

    const int B = in_sizes[0] / DD;          // 8192
    dim3 grid(B / 16), block(32);
    iaf_layer_kernel<<<grid, block, 0, stream>>>(x, W1, b1, Wmu, bmu, Wls, bls, out, B);
}